// MalwareGNN_25237227831713
// MI455X (gfx1250) — compile-verified
//
#include <hip/hip_runtime.h>

typedef __attribute__((ext_vector_type(2))) float v2f;
typedef __attribute__((ext_vector_type(8))) float v8f;

#define HID 64
#define NCLS 5
#define NGR 128

// ---------- norm precompute ----------
__global__ void gcn_deg_init(float* deg, int n) {
    int i = blockIdx.x * blockDim.x + threadIdx.x;
    if (i < n) deg[i] = 1.0f;                     // self-loop
}

__global__ void gcn_deg_count(const int* __restrict__ dst, float* deg, int e) {
    int i = blockIdx.x * blockDim.x + threadIdx.x;
    if (i < e) atomicAdd(&deg[dst[i]], 1.0f);
}

__global__ void gcn_dinv(float* deg, int n) {
    int i = blockIdx.x * blockDim.x + threadIdx.x;
    if (i < n) deg[i] = rsqrtf(deg[i]);           // deg >= 1 always
}

__global__ void gcn_norm(const int* __restrict__ src, const int* __restrict__ dst,
                         const float* __restrict__ dinv, float* nrm, int e) {
    int i = blockIdx.x * blockDim.x + threadIdx.x;
    if (i < e) nrm[i] = dinv[src[i]] * dinv[dst[i]];
}

__global__ void gcn_zero(float* p, long n) {
    long i = (long)blockIdx.x * blockDim.x + threadIdx.x;
    if (i < n) p[i] = 0.0f;
}

// ---------- dense GEMM: H[M,64] = X[M,K] @ W[K,64], fp32 WMMA ----------
// One wave computes a 16-row M-tile against all 4 column tiles (64 cols).
// A layout (16x4 f32): lanes 0-15 = rows with K+0/K+1; lanes 16-31 = K+2/K+3.
// W is staged in LDS K-pair-interleaved: sW[(k>>1)*128 + n*2 + (k&1)] so each
// B operand (W[k][n], W[k+1][n]) is one aligned ds_load_b64 into a VGPR pair.
__global__ __launch_bounds__(256) void gcn_gemm_wmma(const float* __restrict__ X,
                                                     const float* __restrict__ W,
                                                     float* __restrict__ H,
                                                     int M, int K) {
    __shared__ float sW[128 * HID];               // up to 32 KB (K<=128)
    for (int idx = threadIdx.x; idx < K * HID; idx += 256) {
        int k = idx >> 6;                          // row of W
        int n = idx & 63;                          // col of W
        sW[(k >> 1) * 128 + (n << 1) + (k & 1)] = W[idx];
    }
    __syncthreads();

    int wave = threadIdx.x >> 5;
    int lane = threadIdx.x & 31;
    int hh   = lane >> 4;                          // K-half select
    int m    = lane & 15;                          // row-in-tile / col-in-tile
    long row0 = ((long)blockIdx.x * 8 + wave) * 16;
    if (row0 >= M) return;                         // wave-uniform

    v8f a0 = {}, a1 = {}, a2 = {}, a3 = {};
    const float* xrow = X + (row0 + m) * (long)K;

    for (int k = 0; k < K; k += 4) {
        v2f a = *(const v2f*)(xrow + k + 2 * hh);          // 8B global load
        const float* bp = sW + ((k >> 1) + hh) * 128 + (m << 1);
        v2f b0 = *(const v2f*)(bp +  0);                    // ds_load_b64
        v2f b1 = *(const v2f*)(bp + 32);
        v2f b2 = *(const v2f*)(bp + 64);
        v2f b3 = *(const v2f*)(bp + 96);
        a0 = __builtin_amdgcn_wmma_f32_16x16x4_f32(false, a, false, b0, (short)0, a0, false, false);
        a1 = __builtin_amdgcn_wmma_f32_16x16x4_f32(false, a, false, b1, (short)0, a1, false, false);
        a2 = __builtin_amdgcn_wmma_f32_16x16x4_f32(false, a, false, b2, (short)0, a2, false, false);
        a3 = __builtin_amdgcn_wmma_f32_16x16x4_f32(false, a, false, b3, (short)0, a3, false, false);
    }

    // C/D layout: vgpr v holds row (v + 8*hh), column = lane&15 within each 16-col tile
    #pragma unroll
    for (int v = 0; v < 8; v++) {
        float* hp = H + (row0 + v + 8 * hh) * HID + m;
        hp[0]  = a0[v];
        hp[16] = a1[v];
        hp[32] = a2[v];
        hp[48] = a3[v];
    }
}

// ---------- edge scatter: agg[dst] += norm * h[src] ----------
__global__ void gcn_scatter(const int* __restrict__ src, const int* __restrict__ dst,
                            const float* __restrict__ nrm, const float* __restrict__ h,
                            float* agg, long e) {
    long t = (long)blockIdx.x * blockDim.x + threadIdx.x;
    if (t >= e * 16) return;
    long ei = t >> 4;
    int  c  = (int)(t & 15) * 4;
    int  s  = src[ei];
    int  d  = dst[ei];
    float n = nrm[ei];
    float4 v = *(const float4*)(h + (long)s * HID + c);
    float* ap = agg + (long)d * HID + c;
    atomicAdd(ap + 0, n * v.x);
    atomicAdd(ap + 1, n * v.y);
    atomicAdd(ap + 2, n * v.z);
    atomicAdd(ap + 3, n * v.w);
}

// ---------- finalize: out = relu(agg + dinv^2 * h + bias) ----------
__global__ void gcn_finalize(const float* __restrict__ agg, const float* __restrict__ h,
                             const float* __restrict__ dinv, const float* __restrict__ bias,
                             float* out, long n) {
    long t = (long)blockIdx.x * blockDim.x + threadIdx.x;
    if (t >= n * 16) return;
    long i = t >> 4;
    int  c = (int)(t & 15) * 4;
    float dv = dinv[i];
    float sl = dv * dv;
    float4 av = *(const float4*)(agg + i * HID + c);
    float4 hv = *(const float4*)(h   + i * HID + c);
    float4 r;
    r.x = fmaxf(av.x + sl * hv.x + bias[c + 0], 0.0f);
    r.y = fmaxf(av.y + sl * hv.y + bias[c + 1], 0.0f);
    r.z = fmaxf(av.z + sl * hv.z + bias[c + 2], 0.0f);
    r.w = fmaxf(av.w + sl * hv.w + bias[c + 3], 0.0f);
    *(float4*)(out + i * HID + c) = r;
}

// ---------- mean pool ----------
__global__ void gcn_pool(const float* __restrict__ x, const int* __restrict__ batch,
                         float* psum, float* pcnt, long n) {
    long t = (long)blockIdx.x * blockDim.x + threadIdx.x;
    if (t >= n * 16) return;
    long i = t >> 4;
    int  c = (int)(t & 15) * 4;
    int  g = batch[i];
    float4 v = *(const float4*)(x + i * HID + c);
    float* sp = psum + (long)g * HID + c;
    atomicAdd(sp + 0, v.x);
    atomicAdd(sp + 1, v.y);
    atomicAdd(sp + 2, v.z);
    atomicAdd(sp + 3, v.w);
    if ((t & 15) == 0) atomicAdd(&pcnt[g], 1.0f);
}

// ---------- classifier head: out[g,c] = mean(h_g) @ Wc + bc ----------
__global__ void gcn_classify(const float* __restrict__ psum, const float* __restrict__ pcnt,
                             const float* __restrict__ Wc, const float* __restrict__ bc,
                             float* out) {
    int g = threadIdx.x;                          // 128 graphs, one block
    float inv = 1.0f / fmaxf(pcnt[g], 1.0f);
    float acc[NCLS];
    #pragma unroll
    for (int c = 0; c < NCLS; c++) acc[c] = bc[c];
    for (int j = 0; j < HID; j++) {
        float p = psum[g * HID + j] * inv;
        #pragma unroll
        for (int c = 0; c < NCLS; c++) acc[c] += p * Wc[j * NCLS + c];
    }
    #pragma unroll
    for (int c = 0; c < NCLS; c++) out[g * NCLS + c] = acc[c];
}

static inline int cdiv(long a, long b) { return (int)((a + b - 1) / b); }

extern "C" void kernel_launch(void* const* d_in, const int* in_sizes, int n_in,
                              void* d_out, int out_size, void* d_ws, size_t ws_size,
                              hipStream_t stream) {
    const float* x     = (const float*)d_in[0];
    const int*   ei    = (const int*)d_in[1];
    const int*   batch = (const int*)d_in[2];
    const float* W1 = (const float*)d_in[3];  const float* b1 = (const float*)d_in[4];
    const float* W2 = (const float*)d_in[5];  const float* b2 = (const float*)d_in[6];
    const float* W3 = (const float*)d_in[7];  const float* b3 = (const float*)d_in[8];
    const float* Wc = (const float*)d_in[9];  const float* bc = (const float*)d_in[10];
    float* out = (float*)d_out;

    const int N = in_sizes[0] / 128;             // 100000 (multiple of 16)
    const int E = in_sizes[1] / 2;               // 1200000
    const int* src = ei;
    const int* dst = ei + E;

    // workspace layout (floats)
    float* ws   = (float*)d_ws;
    float* deg  = ws;                            // N    (becomes dinv in place)
    float* nrm  = deg  + N;                      // E
    float* hbuf = nrm  + E;                      // N*HID  (GEMM output)
    float* abuf = hbuf + (size_t)N * HID;        // N*HID  (scatter accumulator)
    float* xbuf = abuf + (size_t)N * HID;        // N*HID  (next-layer input)
    float* psum = xbuf + (size_t)N * HID;        // NGR*HID
    float* pcnt = psum + (size_t)NGR * HID;      // NGR

    // --- symmetric normalization ---
    gcn_deg_init <<<cdiv(N, 256), 256, 0, stream>>>(deg, N);
    gcn_deg_count<<<cdiv(E, 256), 256, 0, stream>>>(dst, deg, E);
    gcn_dinv     <<<cdiv(N, 256), 256, 0, stream>>>(deg, N);
    gcn_norm     <<<cdiv(E, 256), 256, 0, stream>>>(src, dst, deg, nrm, E);

    const int gemmBlocks = cdiv(N / 16, 8);
    const int scatBlocks = cdiv((long)E * 16, 256);
    const int nodeBlocks = cdiv((long)N * 16, 256);

    const float* Ws[3] = { W1, W2, W3 };
    const float* Bs[3] = { b1, b2, b3 };
    for (int layer = 0; layer < 3; layer++) {
        const float* Xin = (layer == 0) ? x : xbuf;
        int K = (layer == 0) ? 128 : HID;
        gcn_gemm_wmma<<<gemmBlocks, 256, 0, stream>>>(Xin, Ws[layer], hbuf, N, K);
        gcn_zero     <<<cdiv((long)N * HID, 256), 256, 0, stream>>>(abuf, (long)N * HID);
        gcn_scatter  <<<scatBlocks, 256, 0, stream>>>(src, dst, nrm, hbuf, abuf, E);
        gcn_finalize <<<nodeBlocks, 256, 0, stream>>>(abuf, hbuf, deg, Bs[layer], xbuf, N);
    }

    // --- mean pool + head ---
    gcn_zero    <<<cdiv((long)NGR * HID + NGR, 256), 256, 0, stream>>>(psum, (long)NGR * HID + NGR);
    gcn_pool    <<<nodeBlocks, 256, 0, stream>>>(xbuf, batch, psum, pcnt, N);
    gcn_classify<<<1, NGR, 0, stream>>>(psum, pcnt, Wc, bc, out);
}